// GATLayer_31877247271119
// MI455X (gfx1250) — compile-verified
//
#include <hip/hip_runtime.h>
#include <hip/hip_bf16.h>

#define N_NODES 100000
#define N_EDGES 1600000
#define IN_DIM  256
#define OUT_DIM 64
#define NEG_SLOPE 0.01f

typedef __attribute__((ext_vector_type(2))) float v2f;
typedef __attribute__((ext_vector_type(8))) float v8f;

// ---------------------------------------------------------------------------
// K1: z = h @ W_fc^T via V_WMMA_F32_16X16X4_F32.
// One wave owns a full 16x64 output tile (4 accumulators), so each h row is
// read exactly once from HBM (102.4 MB compulsory traffic) and the A fragment
// is reused across 4 WMMAs per K-step. 6250 waves total.
// A layout (16x4 f32): lanes 0-15 -> M=lane, K={0,1}; lanes 16-31 -> K={2,3}.
// B layout symmetric (rows across lanes). C/D: VGPR i holds rows i / i+8.
// ---------------------------------------------------------------------------
__global__ void __launch_bounds__(256)
gat_gemm_wmma(const float* __restrict__ h, const float* __restrict__ W,
              float* __restrict__ z) {
  const int wave = (blockIdx.x * blockDim.x + threadIdx.x) >> 5;
  const int lane = threadIdx.x & 31;
  if (wave >= N_NODES / 16) return;   // wave-uniform guard (EXEC stays all-1s)
  const int m0   = wave * 16;
  const int half = lane >> 4;         // 0: K=0,1 ; 1: K=2,3
  const int l    = lane & 15;
  const float* arow = h + (size_t)(m0 + l) * IN_DIM + 2 * half;
  const float* b0 = W + (size_t)(l +  0) * IN_DIM + 2 * half;  // B[k][n]=W[n][k]
  const float* b1 = W + (size_t)(l + 16) * IN_DIM + 2 * half;
  const float* b2 = W + (size_t)(l + 32) * IN_DIM + 2 * half;
  const float* b3 = W + (size_t)(l + 48) * IN_DIM + 2 * half;
  v8f c0 = {}, c1 = {}, c2 = {}, c3 = {};
#pragma unroll 2
  for (int k = 0; k < IN_DIM; k += 4) {
    v2f a   = *(const v2f*)(arow + k);
    v2f vb0 = *(const v2f*)(b0 + k);
    v2f vb1 = *(const v2f*)(b1 + k);
    v2f vb2 = *(const v2f*)(b2 + k);
    v2f vb3 = *(const v2f*)(b3 + k);
    c0 = __builtin_amdgcn_wmma_f32_16x16x4_f32(false, a, false, vb0, (short)0, c0, false, false);
    c1 = __builtin_amdgcn_wmma_f32_16x16x4_f32(false, a, false, vb1, (short)0, c1, false, false);
    c2 = __builtin_amdgcn_wmma_f32_16x16x4_f32(false, a, false, vb2, (short)0, c2, false, false);
    c3 = __builtin_amdgcn_wmma_f32_16x16x4_f32(false, a, false, vb3, (short)0, c3, false, false);
  }
  const int rbase = m0 + half * 8;
  float* zb = z + (size_t)rbase * OUT_DIM + l;
#pragma unroll
  for (int i = 0; i < 8; ++i) {
    zb[(size_t)i * OUT_DIM +  0] = c0[i];
    zb[(size_t)i * OUT_DIM + 16] = c1[i];
    zb[(size_t)i * OUT_DIM + 32] = c2[i];
    zb[(size_t)i * OUT_DIM + 48] = c3[i];
  }
}

// ---------------------------------------------------------------------------
// K2: per-node attention scores s_src = z.a_l, s_dst = z.a_r (one wave/node)
// ---------------------------------------------------------------------------
__global__ void __launch_bounds__(256)
gat_scores(const float* __restrict__ z, const float* __restrict__ a_attn,
           float* __restrict__ s_src, float* __restrict__ s_dst) {
  const int node = (blockIdx.x * blockDim.x + threadIdx.x) >> 5;
  const int lane = threadIdx.x & 31;
  if (node >= N_NODES) return;
  const float* zr = z + (size_t)node * OUT_DIM;
  const float z0 = zr[lane], z1 = zr[lane + 32];
  float ssrc = z0 * a_attn[lane]           + z1 * a_attn[lane + 32];
  float sdst = z0 * a_attn[OUT_DIM + lane] + z1 * a_attn[OUT_DIM + lane + 32];
#pragma unroll
  for (int off = 16; off > 0; off >>= 1) {
    ssrc += __shfl_xor(ssrc, off, 32);
    sdst += __shfl_xor(sdst, off, 32);
  }
  if (lane == 0) { s_src[node] = ssrc; s_dst[node] = sdst; }
}

// ---------------------------------------------------------------------------
// K3: init emax=-inf, denom=0, h_out=0
// ---------------------------------------------------------------------------
__global__ void __launch_bounds__(256)
gat_init(float* __restrict__ emax, float* __restrict__ denom,
         float* __restrict__ hout) {
  const size_t i = (size_t)blockIdx.x * blockDim.x + threadIdx.x;
  if (i < N_NODES) { emax[i] = -__builtin_inff(); denom[i] = 0.f; }
  if (i < (size_t)N_NODES * OUT_DIM) hout[i] = 0.f;
}

__device__ __forceinline__ float leaky(float x) {
  return x >= 0.f ? x : NEG_SLOPE * x;
}

// monotone float atomic-max via int/uint punning (init must be -inf)
__device__ __forceinline__ void atomicMaxF(float* a, float v) {
  if (v >= 0.f) atomicMax((int*)a, __float_as_int(v));
  else          atomicMin((unsigned int*)a, __float_as_uint(v));
}

// ---------------------------------------------------------------------------
// K4: segment-max of leaky_relu(s_src[src]+s_dst[dst]) over dst
// ---------------------------------------------------------------------------
__global__ void __launch_bounds__(256)
gat_edge_max(const int* __restrict__ ei, const float* __restrict__ s_src,
             const float* __restrict__ s_dst, float* __restrict__ emax) {
  const int e = blockIdx.x * blockDim.x + threadIdx.x;
  if (e >= N_EDGES) return;
  const int src = ei[e];
  const int dst = ei[N_EDGES + e];
  atomicMaxF(&emax[dst], leaky(s_src[src] + s_dst[dst]));
}

// K5: emax = isfinite(emax) ? emax : 0
__global__ void __launch_bounds__(256)
gat_fix_max(float* __restrict__ emax) {
  const int i = blockIdx.x * blockDim.x + threadIdx.x;
  if (i >= N_NODES) return;
  const float v = emax[i];
  if (!(__builtin_fabsf(v) < __builtin_inff())) emax[i] = 0.f;  // NaN/±inf -> 0
}

// ---------------------------------------------------------------------------
// K6: ex = exp(e - emax[dst]); denom[dst] += ex; stash ex per edge
// ---------------------------------------------------------------------------
__global__ void __launch_bounds__(256)
gat_edge_exp(const int* __restrict__ ei, const float* __restrict__ s_src,
             const float* __restrict__ s_dst, const float* __restrict__ emax,
             float* __restrict__ denom, float* __restrict__ exv) {
  const int e = blockIdx.x * blockDim.x + threadIdx.x;
  if (e >= N_EDGES) return;
  const int src = ei[e];
  const int dst = ei[N_EDGES + e];
  const float v  = leaky(s_src[src] + s_dst[dst]);
  const float ex = __expf(v - emax[dst]);
  exv[e] = ex;
  atomicAdd(&denom[dst], ex);
}

// ---------------------------------------------------------------------------
// K7: h_out[dst] += (ex/denom[dst]) * z[src]; one thread per (edge, channel)
// ---------------------------------------------------------------------------
__global__ void __launch_bounds__(256)
gat_scatter(const int* __restrict__ ei, const float* __restrict__ z,
            const float* __restrict__ exv, const float* __restrict__ denom,
            float* __restrict__ hout) {
  const long long t = (long long)blockIdx.x * blockDim.x + threadIdx.x;
  if (t >= (long long)N_EDGES * OUT_DIM) return;
  const int e = (int)(t >> 6);
  const int j = (int)(t & 63);
  const int src = ei[e];
  const int dst = ei[N_EDGES + e];
  const float d = denom[dst];
  const float alpha = exv[e] / (d > 0.f ? d : 1.f);
  atomicAdd(&hout[(size_t)dst * OUT_DIM + j],
            alpha * z[(size_t)src * OUT_DIM + j]);
}

extern "C" void kernel_launch(void* const* d_in, const int* in_sizes, int n_in,
                              void* d_out, int out_size, void* d_ws, size_t ws_size,
                              hipStream_t stream) {
  const float* h      = (const float*)d_in[0];
  const float* W      = (const float*)d_in[1];
  const float* a_attn = (const float*)d_in[2];
  const int*   ei     = (const int*)d_in[3];
  float* hout = (float*)d_out;

  float* ws    = (float*)d_ws;
  float* z     = ws;                                   // 6,400,000 f32
  float* s_src = z + (size_t)N_NODES * OUT_DIM;        //   100,000
  float* s_dst = s_src + N_NODES;                      //   100,000
  float* emax  = s_dst + N_NODES;                      //   100,000
  float* denom = emax + N_NODES;                       //   100,000
  float* exv   = denom + N_NODES;                      // 1,600,000

  const int waves = N_NODES / 16;                      // 6250 row-tiles
  gat_gemm_wmma<<<(waves + 7) / 8, 256, 0, stream>>>(h, W, z);
  gat_scores<<<(N_NODES + 7) / 8, 256, 0, stream>>>(z, a_attn, s_src, s_dst);

  const int tot = N_NODES * OUT_DIM;
  gat_init<<<(tot + 255) / 256, 256, 0, stream>>>(emax, denom, hout);
  gat_edge_max<<<(N_EDGES + 255) / 256, 256, 0, stream>>>(ei, s_src, s_dst, emax);
  gat_fix_max<<<(N_NODES + 255) / 256, 256, 0, stream>>>(emax);
  gat_edge_exp<<<(N_EDGES + 255) / 256, 256, 0, stream>>>(ei, s_src, s_dst, emax,
                                                          denom, exv);
  const long long tot2 = (long long)N_EDGES * OUT_DIM; // 102,400,000
  gat_scatter<<<(int)((tot2 + 255) / 256), 256, 0, stream>>>(ei, z, exv, denom,
                                                             hout);
}